// GraphPolicyNetwork_36017595744691
// MI455X (gfx1250) — compile-verified
//
#include <hip/hip_runtime.h>
#include <hip/hip_bf16.h>
#include <math.h>

typedef float v2f __attribute__((ext_vector_type(2)));
typedef float v8f __attribute__((ext_vector_type(8)));

#define NNODES 100000
#define NEDGES 1600000
#define DFEAT  64
#define NCLS   4
#define KTOT   128        // combined K: 64 (self) + 64 (neighbor)
#define LSTR   132        // LDS row stride: 16B-aligned float4s, bank-spread

// ---------------------------------------------------------------------------
// Edge scatter: agg[dst] += feat[src] (64 f32 per edge), deg[dst] += 1.
// 16 threads per edge, each handling a float4 chunk (global_atomic_add_f32).
// ---------------------------------------------------------------------------
__global__ void scatter_mean_kernel(const float* __restrict__ feat,
                                    const int* __restrict__ esrc,
                                    const int* __restrict__ edst,
                                    float* __restrict__ agg,
                                    float* __restrict__ deg,
                                    long long total) {
  const long long stride = (long long)gridDim.x * blockDim.x;
  for (long long tid = (long long)blockIdx.x * blockDim.x + threadIdx.x;
       tid < total; tid += stride) {
    const int e = (int)(tid >> 4);
    const int q = (int)(tid & 15);
    const int s = esrc[e];
    const int d = edst[e];
    const float4 v =
        *reinterpret_cast<const float4*>(feat + (long long)s * DFEAT + q * 4);
    float* dst = agg + (long long)d * DFEAT + q * 4;
    atomicAdd(dst + 0, v.x);
    atomicAdd(dst + 1, v.y);
    atomicAdd(dst + 2, v.z);
    atomicAdd(dst + 3, v.w);
    if (q == 0) atomicAdd(deg + d, 1.0f);
  }
}

// invdeg[i] = 1 / max(deg[i], 1)   (in place)
__global__ void invdeg_kernel(float* __restrict__ deg, int n) {
  const int i = blockIdx.x * blockDim.x + threadIdx.x;
  if (i < n) deg[i] = 1.0f / fmaxf(deg[i], 1.0f);
}

// ---------------------------------------------------------------------------
// Fused SAGE layer: out = act( X @ Wself + (agg*invdeg) @ Wneigh + bias )
//
// Block = one 16-node row tile. LDS staging:
//   ldsA[r][k]  r=0..15, k=0..127 :  k<64 -> X[row][k], k>=64 -> aggnorm
//   ldsW[c][k]  c=0..COLS-1       :  transposed [Wself;Wneigh], zero-padded
// Each wave computes a 16x16 output tile with 32 chained
// V_WMMA_F32_16X16X4_F32 ops; inner loop is branch-free (2x ds_load_b64 +
// 1x wmma per K-step).
//
// Fragment layouts (CDNA5 ISA 7.12.2, wave32):
//   A 16x4 f32 : lane L holds row M=L&15, K-pair = 2*(L>>4)       -> v2f
//   B 4x16 f32 : lane L holds col N=L&15, K-pair = 2*(L>>4)       -> v2f
//   C/D 16x16  : element c[r] at lane L is (M = r + 8*(L>>4), N=L&15)
// ---------------------------------------------------------------------------
template <int NOUT, bool ACT>
__global__ __launch_bounds__(128)
void sage_wmma_kernel(const float* __restrict__ X,       // [n, 64]
                      const float* __restrict__ agg,     // [n, 64] neighbor sums
                      const float* __restrict__ invdeg,  // [n]
                      const float* __restrict__ Wself,   // [64, NOUT] row-major
                      const float* __restrict__ Wneigh,  // [64, NOUT] row-major
                      const float* __restrict__ bias,    // [NOUT]
                      float* __restrict__ out,           // [n, NOUT]
                      int nNodes) {
  constexpr int COLS = (NOUT + 15) & ~15;  // 64 (layer1) or 16 (layer2)
  constexpr int CH = NOUT / 4;             // float4 chunks per W row

  __shared__ __align__(16) float ldsA[16 * LSTR];
  __shared__ __align__(16) float ldsW[COLS * LSTR];

  const int nodeBase = blockIdx.x * 16;

  // ---- Stage A: [X | agg*invdeg] tile, coalesced float4 loads ----
  for (int i = threadIdx.x; i < 16 * (KTOT / 4); i += blockDim.x) {
    int r = i >> 5;            // tile row 0..15
    int c4 = (i & 31) << 2;    // k group 0,4,...,124
    int row = nodeBase + r;
    row = row < nNodes ? row : nNodes - 1;
    float4 v;
    if (c4 < DFEAT) {
      v = *reinterpret_cast<const float4*>(X + (long long)row * DFEAT + c4);
    } else {
      v = *reinterpret_cast<const float4*>(agg + (long long)row * DFEAT +
                                           (c4 - DFEAT));
      const float inv = invdeg[row];
      v.x *= inv; v.y *= inv; v.z *= inv; v.w *= inv;
    }
    *reinterpret_cast<float4*>(ldsA + r * LSTR + c4) = v;
  }

  // ---- Stage W transposed: ldsW[c][k] = [Wself;Wneigh][k][c] ----
  for (int i = threadIdx.x; i < KTOT * CH; i += blockDim.x) {
    int k = i / CH;
    int c4 = (i % CH) * 4;
    const float* src = (k < DFEAT) ? (Wself + k * NOUT + c4)
                                   : (Wneigh + (k - DFEAT) * NOUT + c4);
    const float4 w = *reinterpret_cast<const float4*>(src);
    ldsW[(c4 + 0) * LSTR + k] = w.x;
    ldsW[(c4 + 1) * LSTR + k] = w.y;
    ldsW[(c4 + 2) * LSTR + k] = w.z;
    ldsW[(c4 + 3) * LSTR + k] = w.w;
  }
  // Zero-pad columns NOUT..COLS-1 (layer 2 only)
  if (NOUT < COLS) {
    for (int i = threadIdx.x; i < (COLS - NOUT) * KTOT; i += blockDim.x) {
      int c = NOUT + (i >> 7);
      int k = i & (KTOT - 1);
      ldsW[c * LSTR + k] = 0.0f;
    }
  }
  __syncthreads();

  const int lane = threadIdx.x & 31;
  const int wave = threadIdx.x >> 5;
  const int colBase = wave * 16;
  const int nIdx = lane & 15;   // A row / B,C column within tile
  const int kh = lane >> 4;     // K-pair select (A/B), M-half select (C/D)
  const int col = colBase + nIdx;

  v8f acc = {0.f, 0.f, 0.f, 0.f, 0.f, 0.f, 0.f, 0.f};

  // Branch-free MACC loop over combined K = 128
#pragma unroll
  for (int kk = 0; kk < KTOT; kk += 4) {
    const int k0 = kk + 2 * kh;
    const v2f a = *reinterpret_cast<const v2f*>(ldsA + nIdx * LSTR + k0);
    const v2f b = *reinterpret_cast<const v2f*>(ldsW + col * LSTR + k0);
    acc = __builtin_amdgcn_wmma_f32_16x16x4_f32(false, a, false, b, (short)0,
                                                acc, false, false);
  }

  const bool colOK = (col < NOUT);
  const float bcol = colOK ? bias[col] : 0.0f;
#pragma unroll
  for (int r = 0; r < 8; ++r) {
    const int M = r + 8 * kh;
    const int row = nodeBase + M;
    if (row < nNodes && colOK) {
      float v = acc[r] + bcol;
      if (ACT) v = tanhf(v);
      out[(long long)row * NOUT + col] = v;
    }
  }
}

// ---------------------------------------------------------------------------
// Host-side orchestration (all on `stream`, graph-capture safe).
// Workspace layout: [ agg : N*64 f32 | deg : N f32 | h1 : N*64 f32 ]
// ---------------------------------------------------------------------------
extern "C" void kernel_launch(void* const* d_in, const int* in_sizes, int n_in,
                              void* d_out, int out_size, void* d_ws,
                              size_t ws_size, hipStream_t stream) {
  (void)in_sizes; (void)n_in; (void)out_size; (void)ws_size;

  const float* X    = (const float*)d_in[0];  // node_features [N,64]
  const int*   esrc = (const int*)  d_in[1];  // edge_src
  const int*   edst = (const int*)  d_in[2];  // edge_dst
  const float* Ws1  = (const float*)d_in[3];  // W_self1 [64,64]
  const float* bs1  = (const float*)d_in[4];  // b_self1 [64]
  const float* Wn1  = (const float*)d_in[5];  // W_neigh1 [64,64]
  const float* Ws2  = (const float*)d_in[6];  // W_self2 [64,4]
  const float* bs2  = (const float*)d_in[7];  // b_self2 [4]
  const float* Wn2  = (const float*)d_in[8];  // W_neigh2 [64,4]
  float* out = (float*)d_out;

  float* agg = (float*)d_ws;
  float* deg = agg + (size_t)NNODES * DFEAT;
  float* h1  = deg + NNODES;

  const long long scatterWork = (long long)NEDGES * (DFEAT / 4);
  const int nodeBlocks = (NNODES + 15) / 16;

  // ---- Layer 1 ----
  hipMemsetAsync(agg, 0, (size_t)(NNODES * DFEAT + NNODES) * sizeof(float),
                 stream);
  scatter_mean_kernel<<<2048, 256, 0, stream>>>(X, esrc, edst, agg, deg,
                                                scatterWork);
  invdeg_kernel<<<(NNODES + 255) / 256, 256, 0, stream>>>(deg, NNODES);
  sage_wmma_kernel<DFEAT, true><<<nodeBlocks, 128, 0, stream>>>(
      X, agg, deg, Ws1, Wn1, bs1, h1, NNODES);

  // ---- Layer 2 ----
  hipMemsetAsync(agg, 0, (size_t)(NNODES * DFEAT + NNODES) * sizeof(float),
                 stream);
  scatter_mean_kernel<<<2048, 256, 0, stream>>>(h1, esrc, edst, agg, deg,
                                                scatterWork);
  invdeg_kernel<<<(NNODES + 255) / 256, 256, 0, stream>>>(deg, NNODES);
  sage_wmma_kernel<NCLS, false><<<nodeBlocks, 32, 0, stream>>>(
      h1, agg, deg, Ws2, Wn2, bs2, out, NNODES);
}